// DepthProb_43044162241049
// MI455X (gfx1250) — compile-verified
//
#include <hip/hip_runtime.h>
#include <hip/hip_bf16.h>
#include <stdint.h>

#define KBINS 64
#define HW_SHIFT 19                 // H*W = 512*1024 = 2^19
#define HW (1 << HW_SHIFT)
#define LOG2E 1.4426950408889634f

// -------------------------------------------------------------------------
// DepthProb: out[b,k,h,w] = softmax_k( -(feat[b,0,h,w] - depint[k])^2 )
//
// Memory-bound: 512MB of output stores (22us floor @ 23.3 TB/s) vs ~0.6
// GFLOP + 134M v_exp_f32. Strategy: 1 pixel/thread, depint staged to LDS
// via gfx1250 async global->LDS load (ASYNCcnt path), register-resident
// 64-bin softmax in exp2 domain, 64 coalesced nontemporal stores (output
// >> 192MB L2). 4-way split accumulators keep min/sum chains short and
// v_pk_* friendly; float4 LDS reads force ds_load_b128.
// -------------------------------------------------------------------------

#if defined(__gfx1250__) && __has_builtin(__builtin_amdgcn_global_load_async_to_lds_b32)
typedef __attribute__((address_space(1))) int* gptr_i32;
typedef __attribute__((address_space(3))) int* lptr_i32;
#define HAVE_ASYNC_LDS 1
#endif

#if __has_builtin(__builtin_amdgcn_exp2f)
#define HAVE_EXP2 1
#endif

__global__ __launch_bounds__(256) void DepthProb_kernel(
    const float* __restrict__ feat,
    const float* __restrict__ depint,
    float* __restrict__ out,
    int npix)
{
    __shared__ __align__(16) float sdep[KBINS];
    const int t = threadIdx.x;

#ifdef HAVE_ASYNC_LDS
    // CDNA5 async copy: per-lane async load of depint[t] into LDS, tracked
    // by ASYNCcnt (not LOADcnt). Lanes >= 64 are EXEC-masked off.
    if (t < KBINS) {
        gptr_i32 gsrc = (gptr_i32)(uintptr_t)(depint + t);
        lptr_i32 ldst = (lptr_i32)(uint32_t)(uintptr_t)(&sdep[t]);
        __builtin_amdgcn_global_load_async_to_lds_b32(gsrc, ldst, 0, 0);
    }
    asm volatile("s_wait_asynccnt 0x0" ::: "memory");
#else
    if (t < KBINS) sdep[t] = depint[t];
#endif
    __syncthreads();

    const int p = blockIdx.x * 256 + t;
    if (p >= npix) return;

    const float f = feat[p];

    // Pass 1: squared distances to all 64 bins; float4 LDS reads
    // (ds_load_b128) + 4-way split min (short chains, packable).
    const float4* sd4 = (const float4*)sdep;
    float t2[KBINS];
    float m0 = 3.4028235e38f, m1 = m0, m2 = m0, m3 = m0;
#pragma unroll
    for (int j = 0; j < KBINS / 4; ++j) {
        const float4 d4 = sd4[j];
        const float d0 = f - d4.x;
        const float d1 = f - d4.y;
        const float d2 = f - d4.z;
        const float d3 = f - d4.w;
        const float v0 = d0 * d0, v1 = d1 * d1, v2 = d2 * d2, v3 = d3 * d3;
        t2[4 * j + 0] = v0; t2[4 * j + 1] = v1;
        t2[4 * j + 2] = v2; t2[4 * j + 3] = v3;
        m0 = fminf(m0, v0); m1 = fminf(m1, v1);
        m2 = fminf(m2, v2); m3 = fminf(m3, v3);
    }
    const float mn = fminf(fminf(m0, m1), fminf(m2, m3));

    // Pass 2: stable exp in exp2 domain: exp(mn - t2) = exp2(fma(t2,-log2e,mnl)).
    // One v_exp_f32 per output element; 4-way split sum.
    float s0 = 0.f, s1 = 0.f, s2 = 0.f, s3 = 0.f;
#ifdef HAVE_EXP2
    const float mnl = mn * LOG2E;
#pragma unroll
    for (int k = 0; k < KBINS; k += 4) {
        const float e0 = __builtin_amdgcn_exp2f(fmaf(t2[k + 0], -LOG2E, mnl));
        const float e1 = __builtin_amdgcn_exp2f(fmaf(t2[k + 1], -LOG2E, mnl));
        const float e2 = __builtin_amdgcn_exp2f(fmaf(t2[k + 2], -LOG2E, mnl));
        const float e3 = __builtin_amdgcn_exp2f(fmaf(t2[k + 3], -LOG2E, mnl));
        t2[k + 0] = e0; t2[k + 1] = e1; t2[k + 2] = e2; t2[k + 3] = e3;
        s0 += e0; s1 += e1; s2 += e2; s3 += e3;
    }
#else
#pragma unroll
    for (int k = 0; k < KBINS; k += 4) {
        const float e0 = __expf(mn - t2[k + 0]);
        const float e1 = __expf(mn - t2[k + 1]);
        const float e2 = __expf(mn - t2[k + 2]);
        const float e3 = __expf(mn - t2[k + 3]);
        t2[k + 0] = e0; t2[k + 1] = e1; t2[k + 2] = e2; t2[k + 3] = e3;
        s0 += e0; s1 += e1; s2 += e2; s3 += e3;
    }
#endif
    const float sum = (s0 + s1) + (s2 + s3);

#if __has_builtin(__builtin_amdgcn_rcpf)
    const float r = __builtin_amdgcn_rcpf(sum);
#else
    const float r = 1.0f / sum;
#endif

    // Pass 3: normalize + 64 coalesced nontemporal stores.
    // out index = b*K*HW + k*HW + hw  =  p + b*(K-1)*HW + k*HW
    const int b = p >> HW_SHIFT;
    float* o = out + p + b * ((KBINS - 1) * HW);
#pragma unroll
    for (int k = 0; k < KBINS; ++k) {
        __builtin_nontemporal_store(t2[k] * r, o + k * HW);
    }
}

extern "C" void kernel_launch(void* const* d_in, const int* in_sizes, int n_in,
                              void* d_out, int out_size, void* d_ws, size_t ws_size,
                              hipStream_t stream)
{
    const float* feat   = (const float*)d_in[0];   // (4,1,512,1024) fp32
    const float* depint = (const float*)d_in[1];   // (64,) fp32
    float* out = (float*)d_out;                    // (4,64,512,1024) fp32

    const int npix = in_sizes[0];                  // B*H*W = 2,097,152
    const int threads = 256;
    const int blocks = (npix + threads - 1) / threads;

    DepthProb_kernel<<<blocks, threads, 0, stream>>>(feat, depint, out, npix);
}